// GCN_77051713290303
// MI455X (gfx1250) — compile-verified
//
#include <hip/hip_runtime.h>
#include <hip/hip_bf16.h>

typedef float v2f __attribute__((ext_vector_type(2)));
typedef float v8f __attribute__((ext_vector_type(8)));

#define NTOT   65536
#define NGR    64
#define NPG    1024
#define C      64
#define OUTC   50
#define KKEEP  820   // ceil(0.8 * 1024)

// ---------------------------------------------------------------------------
// Edge scatter: one wave (32 lanes) per edge; each lane atomically adds 2 of
// the 64 source features into agg[dst]; lane 0 bumps cnt[dst].
// ---------------------------------------------------------------------------
__global__ __launch_bounds__(256) void scatter_edges(
    const float* __restrict__ x, const int* __restrict__ esrc,
    const int* __restrict__ edst, float* __restrict__ agg,
    float* __restrict__ cnt, int E)
{
    int gid  = blockIdx.x * blockDim.x + threadIdx.x;
    int e    = gid >> 5;
    int lane = gid & 31;
    if (e >= E) return;
    int s = esrc[e];
    int d = edst[e];
    const float* xs = x   + (size_t)s * C;
    float*       ad = agg + (size_t)d * C;
    atomicAdd(&ad[lane * 2 + 0], xs[lane * 2 + 0]);
    atomicAdd(&ad[lane * 2 + 1], xs[lane * 2 + 1]);
    if (lane == 0) atomicAdd(&cnt[d], 1.0f);
}

// ---------------------------------------------------------------------------
// SAGE GEMM via f32 WMMA: out = act( [agg/cnt | x] (32x128) @ [Wl ; Wr] + b ).
// [Wl;Wr] (32 KB) is staged once per block into LDS as K-paired v2f so every
// B fragment is one ds_load_b64 shared by all 8 waves.  One wave -> 32 rows
// x 64 cols: 2 A fragments x 4 B tiles x 32 K-steps = 256
// V_WMMA_F32_16X16X4_F32 per wave.  Uniform control flow (EXEC all ones).
// ---------------------------------------------------------------------------
__global__ __launch_bounds__(256) void sage_gemm(
    const float* __restrict__ agg, const float* __restrict__ cnt,
    const float* __restrict__ x,
    const float* __restrict__ Wl,  const float* __restrict__ Wr,
    const float* __restrict__ bias, float* __restrict__ out, int relu)
{
    // sW[k2*64 + n] = { W[2*k2][n], W[2*k2+1][n] }, combined K = 0..127
    __shared__ v2f sW[64 * C];

    for (int i = threadIdx.x; i < 64 * C; i += 256) {   // uniform trip count
        int k2 = i >> 6, n = i & (C - 1);
        int k  = 2 * k2;
        const float* Wsrc = (k < C) ? (Wl + (size_t)k * C)
                                    : (Wr + (size_t)(k - C) * C);
        v2f w; w.x = Wsrc[n]; w.y = Wsrc[C + n];
        sW[i] = w;
    }
    __syncthreads();

    const int lane = threadIdx.x & 31;
    const int wave = (blockIdx.x * blockDim.x + threadIdx.x) >> 5;
    const int m0   = wave * 32;          // 32 rows per wave

    const int arow = lane & 15;          // row within a 16-row tile
    const int koff = (lane >> 4) * 2;    // lanes 16-31 carry K+2 / K+3

    const int r0 = m0 + arow, r1 = m0 + 16 + arow;
    const float* aggR0 = agg + (size_t)r0 * C;
    const float* aggR1 = agg + (size_t)r1 * C;
    const float* xR0   = x   + (size_t)r0 * C;
    const float* xR1   = x   + (size_t)r1 * C;
    const float  inv0  = 1.0f / fmaxf(cnt[r0], 1.0f);
    const float  inv1  = 1.0f / fmaxf(cnt[r1], 1.0f);

    __builtin_prefetch(xR0, 0, 0);       // global_prefetch_b8 on the A stream
    __builtin_prefetch(xR1, 0, 0);

    v8f accA[4] = {{}, {}, {}, {}};      // rows m0..m0+15
    v8f accB[4] = {{}, {}, {}, {}};      // rows m0+16..m0+31

    for (int s = 0; s < 32; ++s) {       // combined K = 128, 4 per step
        const int   kb     = (s & 15) * 4;
        const bool  firstW = (s < 16);
        const float* A0    = firstW ? aggR0 : xR0;
        const float* A1    = firstW ? aggR1 : xR1;
        const float  sc0   = firstW ? inv0  : 1.0f;
        const float  sc1   = firstW ? inv1  : 1.0f;
        const int    kc    = kb + (firstW ? 0 : C);   // combined K base

        v2f a0, a1;
        a0.x = A0[kb + koff + 0] * sc0;  a0.y = A0[kb + koff + 1] * sc0;
        a1.x = A1[kb + koff + 0] * sc1;  a1.y = A1[kb + koff + 1] * sc1;

        const v2f* wrow = &sW[((kc + koff) >> 1) * C + arow];
        v2f b0 = wrow[0], b1 = wrow[16], b2 = wrow[32], b3 = wrow[48];

        accA[0] = __builtin_amdgcn_wmma_f32_16x16x4_f32(false, a0, false, b0, (short)0, accA[0], false, false);
        accB[0] = __builtin_amdgcn_wmma_f32_16x16x4_f32(false, a1, false, b0, (short)0, accB[0], false, false);
        accA[1] = __builtin_amdgcn_wmma_f32_16x16x4_f32(false, a0, false, b1, (short)0, accA[1], false, false);
        accB[1] = __builtin_amdgcn_wmma_f32_16x16x4_f32(false, a1, false, b1, (short)0, accB[1], false, false);
        accA[2] = __builtin_amdgcn_wmma_f32_16x16x4_f32(false, a0, false, b2, (short)0, accA[2], false, false);
        accB[2] = __builtin_amdgcn_wmma_f32_16x16x4_f32(false, a1, false, b2, (short)0, accB[2], false, false);
        accA[3] = __builtin_amdgcn_wmma_f32_16x16x4_f32(false, a0, false, b3, (short)0, accA[3], false, false);
        accB[3] = __builtin_amdgcn_wmma_f32_16x16x4_f32(false, a1, false, b3, (short)0, accB[3], false, false);
    }

    // D layout: VGPR v -> lanes 0-15 hold M=v, lanes 16-31 hold M=v+8.
    const int rbase = (lane >> 4) * 8;
    #pragma unroll
    for (int t = 0; t < 4; ++t) {
        const int   col  = t * 16 + arow;
        const float bcol = bias[col];
        #pragma unroll
        for (int v = 0; v < 8; ++v) {
            float vA = accA[t][v] + bcol;
            float vB = accB[t][v] + bcol;
            if (relu) { vA = fmaxf(vA, 0.0f); vB = fmaxf(vB, 0.0f); }
            out[(size_t)(m0 + rbase + v) * C + col]      = vA;
            out[(size_t)(m0 + 16 + rbase + v) * C + col] = vB;
        }
    }
}

// ---------------------------------------------------------------------------
// score = tanh(h . pool_w / ||pool_w||), one thread per node.
// ---------------------------------------------------------------------------
__global__ __launch_bounds__(256) void score_kernel(
    const float* __restrict__ h, const float* __restrict__ pw,
    float* __restrict__ score)
{
    int n = blockIdx.x * blockDim.x + threadIdx.x;
    float nrm = 0.0f;
    #pragma unroll
    for (int i = 0; i < C; ++i) nrm += pw[i] * pw[i];
    nrm = sqrtf(nrm);
    float dot = 0.0f;
    const float* hr = h + (size_t)n * C;
    #pragma unroll
    for (int i = 0; i < C; ++i) dot += hr[i] * pw[i];
    score[n] = tanhf(dot / nrm);
}

// ---------------------------------------------------------------------------
// Top-k pool: one 1024-thread block per graph. Stable rank-based selection
// (exact top_k semantics, ties -> lower index), then deterministic chunked
// reduction for weighted mean and max. feats[g] = [mean(64) | max(64)].
// ---------------------------------------------------------------------------
__global__ __launch_bounds__(1024) void pool_kernel(
    const float* __restrict__ h, const float* __restrict__ score,
    float* __restrict__ feats)
{
    __shared__ float s_sc[NPG];
    __shared__ int   s_flag[NPG];
    __shared__ float s_sum[16][C];
    __shared__ float s_max[16][C];

    const int g = blockIdx.x;
    const int t = threadIdx.x;

    s_sc[t] = score[g * NPG + t];
    __syncthreads();

    const float my = s_sc[t];
    int rank = 0;
    for (int j = 0; j < NPG; ++j) {
        float sj = s_sc[j];
        rank += (sj > my) || (sj == my && j < t);
    }
    s_flag[t] = (rank < KKEEP) ? 1 : 0;
    __syncthreads();

    // thread t -> feature f over a 64-node chunk c
    const int f = t & (C - 1);
    const int c = t >> 6;
    float sum = 0.0f, mx = -INFINITY;
    for (int j = 0; j < 64; ++j) {
        int nn = c * 64 + j;
        if (s_flag[nn]) {
            float v = h[(size_t)(g * NPG + nn) * C + f] * s_sc[nn];
            sum += v;
            mx = fmaxf(mx, v);
        }
    }
    s_sum[c][f] = sum;
    s_max[c][f] = mx;
    __syncthreads();

    if (t < C) {
        float s2 = 0.0f, m2 = -INFINITY;
        #pragma unroll
        for (int cc = 0; cc < 16; ++cc) {
            s2 += s_sum[cc][t];
            m2 = fmaxf(m2, s_max[cc][t]);
        }
        feats[g * 2 * C + t]     = s2 / (float)KKEEP;
        feats[g * 2 * C + C + t] = m2;
    }
}

// ---------------------------------------------------------------------------
// Head: out[64][50] = feats[64][128] @ lin_W[128][50] + lin_b (tiny).
// ---------------------------------------------------------------------------
__global__ __launch_bounds__(64) void final_linear(
    const float* __restrict__ feats, const float* __restrict__ W,
    const float* __restrict__ b, float* __restrict__ out)
{
    int g = blockIdx.x, o = threadIdx.x;
    if (o >= OUTC) return;
    float acc = b[o];
    const float* fr = feats + g * 2 * C;
    #pragma unroll
    for (int i = 0; i < 2 * C; ++i) acc += fr[i] * W[i * OUTC + o];
    out[g * OUTC + o] = acc;
}

extern "C" void kernel_launch(void* const* d_in, const int* in_sizes, int n_in,
                              void* d_out, int out_size, void* d_ws, size_t ws_size,
                              hipStream_t stream)
{
    (void)n_in; (void)out_size; (void)ws_size;
    const float* x     = (const float*)d_in[0];
    const int*   esrc  = (const int*)  d_in[1];
    const int*   edst  = (const int*)  d_in[2];
    // d_in[3]=num_graphs, d_in[4]=nodes_per_graph (compile-time constants here)
    const float* W1l   = (const float*)d_in[5];
    const float* b1    = (const float*)d_in[6];
    const float* W1r   = (const float*)d_in[7];
    const float* W2l   = (const float*)d_in[8];
    const float* b2    = (const float*)d_in[9];
    const float* W2r   = (const float*)d_in[10];
    const float* pw    = (const float*)d_in[11];
    const float* linW  = (const float*)d_in[12];
    const float* linB  = (const float*)d_in[13];
    float* out = (float*)d_out;

    const int E = in_sizes[1];

    // workspace carve-up (all 256B aligned)
    char* ws = (char*)d_ws;
    float* agg   = (float*)ws;                              // 16 MB
    float* cnt   = (float*)(ws + (size_t)NTOT * C * 4);     // 256 KB
    float* h1    = (float*)(ws + (size_t)NTOT * C * 4 + (size_t)NTOT * 4);
    float* h2    = h1 + (size_t)NTOT * C;                   // 16 MB each
    float* score = h2 + (size_t)NTOT * C;
    float* feats = score + NTOT;

    const int scatterBlocks = (E * 32 + 255) / 256;
    const int gemmBlocks    = NTOT / 32 / 8;   // 8 waves/block, 32 rows/wave

    // ---- conv 1 ----
    hipMemsetAsync(agg, 0, (size_t)NTOT * C * 4, stream);
    hipMemsetAsync(cnt, 0, (size_t)NTOT * 4, stream);
    scatter_edges<<<scatterBlocks, 256, 0, stream>>>(x, esrc, edst, agg, cnt, E);
    sage_gemm<<<gemmBlocks, 256, 0, stream>>>(agg, cnt, x, W1l, W1r, b1, h1, 1);

    // ---- conv 2 ----
    hipMemsetAsync(agg, 0, (size_t)NTOT * C * 4, stream);
    hipMemsetAsync(cnt, 0, (size_t)NTOT * 4, stream);
    scatter_edges<<<scatterBlocks, 256, 0, stream>>>(h1, esrc, edst, agg, cnt, E);
    sage_gemm<<<gemmBlocks, 256, 0, stream>>>(agg, cnt, h1, W2l, W2r, b2, h2, 1);

    // ---- pooling + head ----
    score_kernel<<<NTOT / 256, 256, 0, stream>>>(h2, pw, score);
    pool_kernel<<<NGR, NPG, 0, stream>>>(h2, score, feats);
    final_linear<<<NGR, 64, 0, stream>>>(feats, linW, linB, out);
}